// SpikingLayer_41497974014109
// MI455X (gfx1250) — compile-verified
//
#include <hip/hip_runtime.h>

#define NN 8192
#define TT 64
#define DECAY 0.9f
#define THRESH 1.0f

typedef __attribute__((ext_vector_type(2))) float v2f;
typedef __attribute__((ext_vector_type(8))) float v8f;

// Copy initial state into workspace (run once per kernel_launch call).
__global__ void snn_init_kernel(const float* __restrict__ f0,
                                const float* __restrict__ v0,
                                float* __restrict__ f,
                                float* __restrict__ v, int n) {
    int i = blockIdx.x * blockDim.x + threadIdx.x;
    if (i < n) {
        f[i] = f0[i];
        v[i] = v0[i];
    }
}

// One LIF timestep: cur = W @ f_in (fp32 WMMA matvec), then threshold/reset.
// Block = 256 threads (8 waves). Block b owns rows [16b, 16b+16).
// Wave w covers k in [w*1024, (w+1)*1024) with V_WMMA_F32_16X16X4_F32.
__global__ void __launch_bounds__(256)
snn_step_kernel(const float* __restrict__ W,
                const float* __restrict__ f_in,
                float* __restrict__ f_out,
                float* __restrict__ v,
                float* __restrict__ out,   // [2, T, N]
                int t) {
    __shared__ float partial[8][16];

    const int lane = threadIdx.x & 31;
    const int wave = threadIdx.x >> 5;
    const int r0   = blockIdx.x * 16;

    // A-matrix addressing per ISA layout (32-bit A 16x4):
    //  lane L<16  -> row r0+L,      K = k0+0, k0+1
    //  lane L>=16 -> row r0+(L-16), K = k0+2, k0+3
    const int row  = r0 + (lane & 15);
    const int koff = (lane >> 4) * 2;
    const size_t rowBase = (size_t)row * NN;

    const int kStart = wave * (NN / 8);
    const int kEnd   = kStart + (NN / 8);

    v8f c = {};  // 16x16 f32 accumulator; all columns identical (B replicated)

#pragma unroll 8
    for (int k0 = kStart; k0 < kEnd; k0 += 4) {
        v2f a = *(const v2f*)(W + rowBase + (size_t)(k0 + koff));
        v2f b = *(const v2f*)(f_in + k0 + koff);
        // (neg_a, A, neg_b, B, c_mod, C, reuse_a, reuse_b)
        c = __builtin_amdgcn_wmma_f32_16x16x4_f32(
            false, a, false, b, (short)0, c, false, false);
    }

    // D layout: VGPR r, lane 0 -> (M=r, N=0); lane 16 -> (M=8+r, N=0).
    // All N columns equal, so lanes 0 and 16 hold the 16 partial row sums.
    if ((lane & 15) == 0) {
        const int half = (lane >> 4) * 8;
        partial[wave][half + 0] = c[0];
        partial[wave][half + 1] = c[1];
        partial[wave][half + 2] = c[2];
        partial[wave][half + 3] = c[3];
        partial[wave][half + 4] = c[4];
        partial[wave][half + 5] = c[5];
        partial[wave][half + 6] = c[6];
        partial[wave][half + 7] = c[7];
    }
    __syncthreads();

    // Threads 0..15: reduce the 8 wave-partials and run the LIF update.
    if (threadIdx.x < 16) {
        float cur = 0.0f;
#pragma unroll
        for (int w = 0; w < 8; ++w) cur += partial[w][threadIdx.x];

        const int r = r0 + threadIdx.x;
        const float vnew = DECAY * v[r] + cur;
        const float fire = (vnew >= THRESH) ? 1.0f : 0.0f;

        out[(size_t)t * NN + r]                      = fire;  // fires[t, r]
        out[(size_t)TT * NN + (size_t)t * NN + r]    = vnew;  // vs[t, r]
        f_out[r] = fire;
        v[r]     = (fire > 0.0f) ? 0.0f : vnew;
    }
}

extern "C" void kernel_launch(void* const* d_in, const int* in_sizes, int n_in,
                              void* d_out, int out_size, void* d_ws, size_t ws_size,
                              hipStream_t stream) {
    // Inputs (setup_inputs order): x [T,N] (unused), W [N,N], f0 [N], v0 [N].
    const float* W  = (const float*)d_in[1];
    const float* f0 = (const float*)d_in[2];
    const float* v0 = (const float*)d_in[3];
    float* out = (float*)d_out;

    // Workspace: v state, double-buffered firing vectors (96 KB total).
    float* ws = (float*)d_ws;
    float* v  = ws;
    float* fA = ws + NN;
    float* fB = ws + 2 * NN;

    snn_init_kernel<<<(NN + 255) / 256, 256, 0, stream>>>(f0, v0, fA, v, NN);

    float* fin = fA;
    float* fout = fB;
    for (int t = 0; t < TT; ++t) {
        snn_step_kernel<<<NN / 16, 256, 0, stream>>>(W, fin, fout, v, out, t);
        float* tmp = fin; fin = fout; fout = tmp;
    }
}